// FiBiNet_30279519437134
// MI455X (gfx1250) — compile-verified
//
#include <hip/hip_runtime.h>

#define B_      1024
#define F_      32
#define D_      64
#define P_      496     // F*(F-1)/2
#define OUT_    512
#define EMB_    2048    // F*D
#define RED_    32
#define ZDIM_   128     // F*2*g
#define OSTRIDE_ 2560   // EMB_ + OUT_
#define LN_EPS_ 1e-6f

typedef float v2f __attribute__((ext_vector_type(2)));
typedef float v4f __attribute__((ext_vector_type(4)));
typedef float v8f __attribute__((ext_vector_type(8)));
typedef unsigned int u32x4 __attribute__((ext_vector_type(4)));
typedef int i32x4 __attribute__((ext_vector_type(4)));
typedef int i32x8 __attribute__((ext_vector_type(8)));

// ---------------------------------------------------------------------------
// Kernel 1: SENET squeeze-excite + residual scale + LayerNorm -> out[:, :2048]
// One 256-thread block (8 waves) per batch row.
// ---------------------------------------------------------------------------
__global__ __launch_bounds__(256) void senet_ln_kernel(
    const float* __restrict__ x, const float* __restrict__ W1,
    const float* __restrict__ b1, const float* __restrict__ W2,
    const float* __restrict__ b2, const float* __restrict__ ln_scale,
    const float* __restrict__ ln_bias, float* __restrict__ out)
{
    __shared__ float s_x[EMB_];
    __shared__ float s_z[ZDIM_];
    __shared__ float s_a1[RED_];
    __shared__ float s_red[256];
    __shared__ float s_red2[256];

    const int b = blockIdx.x;
    const int t = threadIdx.x;
    const float* __restrict__ xrow = x + (size_t)b * EMB_;

    #pragma unroll
    for (int c = t * 4; c < EMB_; c += 256 * 4)
        *(v4f*)&s_x[c] = *(const v4f*)&xrow[c];
    __syncthreads();

    // squeeze: per (f, group) max + mean; z layout per f: [mx0, mx1, mn0, mn1]
    if (t < 64) {
        const int f = t >> 1, gg = t & 1;
        const float* seg = &s_x[f * D_ + gg * 32];
        float mx = seg[0], sm = 0.f;
        #pragma unroll
        for (int k = 0; k < 32; ++k) { float v = seg[k]; mx = fmaxf(mx, v); sm += v; }
        s_z[f * 4 + gg]     = mx;
        s_z[f * 4 + 2 + gg] = sm * (1.0f / 32.0f);
    }
    __syncthreads();

    // a1 = relu(z @ W1 + b1), W1 is (128,32) row-major
    if (t < RED_) {
        float acc = b1[t];
        #pragma unroll 8
        for (int k = 0; k < ZDIM_; ++k) acc += s_z[k] * W1[k * RED_ + t];
        s_a1[t] = fmaxf(acc, 0.f);
    }
    __syncthreads();

    // weights = a1 @ W2 + b2 ; senet = flat*weights + flat ; then LayerNorm
    float vals[EMB_ / 256];
    float sum = 0.f, sumsq = 0.f;
    #pragma unroll
    for (int ii = 0; ii < EMB_ / 256; ++ii) {
        const int c = t + ii * 256;
        float w = b2[c];
        #pragma unroll
        for (int r = 0; r < RED_; ++r) w += s_a1[r] * W2[r * EMB_ + c];
        const float v = s_x[c] * w + s_x[c];
        vals[ii] = v;
        sum += v; sumsq += v * v;
    }
    s_red[t] = sum; s_red2[t] = sumsq;
    __syncthreads();
    for (int stride = 128; stride > 0; stride >>= 1) {
        if (t < stride) { s_red[t] += s_red[t + stride]; s_red2[t] += s_red2[t + stride]; }
        __syncthreads();
    }
    const float mu  = s_red[0]  * (1.0f / EMB_);
    const float var = s_red2[0] * (1.0f / EMB_) - mu * mu;
    const float inv = rsqrtf(var + LN_EPS_);

    float* __restrict__ orow = out + (size_t)b * OSTRIDE_;
    #pragma unroll
    for (int ii = 0; ii < EMB_ / 256; ++ii) {
        const int c = t + ii * 256;
        orow[c] = (vals[ii] - mu) * inv * ln_scale[c] + ln_bias[c];
    }
}

// ---------------------------------------------------------------------------
// TDM: async-load a 64x64 f32 row-major tile (16 KB) from global into LDS.
// D# per CDNA5 ISA ch.8: group0 = {count, lds_addr, global_addr, type=2},
// group1 = {data_size=4B, tensor_dim0/1=64, tile_dim0/1=64, dim0_stride=64},
// groups 2/3 zero (3D+ fields unused, tile_dim2=0). Tracked by TENSORcnt.
// This toolchain exposes the 6-arg builtin (extra i32x8 group before cpol).
// ---------------------------------------------------------------------------
__device__ __forceinline__ void tdm_load_tile_64x64_f32(
    float* lds_dst, const float* __restrict__ gsrc)
{
    const unsigned long long ga = (unsigned long long)(uintptr_t)gsrc;
    // LDS aperture flat address: low 32 bits == LDS byte offset (ISA 10.2)
    const unsigned int ldsoff = (unsigned int)(uintptr_t)lds_dst;

    u32x4 g0;
    g0[0] = 1u;                                   // count=1, user descriptor
    g0[1] = ldsoff;                               // lds_addr (bytes)
    g0[2] = (unsigned int)ga;                     // global_addr[31:0]
    g0[3] = (unsigned int)((ga >> 32) & 0x01FFFFFFu) | (2u << 30); // [56:32] | type=2

    i32x8 g1;
    g1[0] = (2 << 16);          // workgroup_mask=0, data_size=2 (4 bytes)
    g1[1] = (int)(64u << 16);   // tensor_dim0 = 64  (bits 79:48, low half here)
    g1[2] = (int)(64u << 16);   // tensor_dim1 = 64  (bits 111:80, low half here)
    g1[3] = (int)(64u << 16);   // tile_dim0   = 64  (bits 127:112)
    g1[4] = 64;                 // tile_dim1   = 64  (bits 143:128); tile_dim2=0
    g1[5] = 64;                 // tensor_dim0_stride = 64 (bits 207:160, low 32)
    g1[6] = 0;
    g1[7] = 0;

    const i32x4 z4 = {0, 0, 0, 0};
    const i32x8 z8 = {0, 0, 0, 0, 0, 0, 0, 0};
    __builtin_amdgcn_tensor_load_to_lds(g0, g1, z4, z4, z8, 0);
}

// ---------------------------------------------------------------------------
// Per-pair 16-row tile: proj = xi(16x64) @ Wp[p](64x64) via fp32 WMMA (K=4),
// then dots[b,p] = sum_e (proj+bp) * xj with a 16-lane xor-shuffle reduction.
// ---------------------------------------------------------------------------
__device__ __forceinline__ void pair_tile_compute(
    const float* __restrict__ x, const float* sW,
    const float* __restrict__ bp, float* __restrict__ dots,
    int p, int i, int j, int mbase, int half, int l)
{
    const v8f vzero = {0.f, 0.f, 0.f, 0.f, 0.f, 0.f, 0.f, 0.f};
    v8f acc[4];
    #pragma unroll
    for (int n = 0; n < 4; ++n) acc[n] = vzero;

    // A (16x4 f32): lanes 0-15 K={0,1}, lanes 16-31 K={2,3}; B mirrors.
    const float* __restrict__ arow = x + ((size_t)(mbase + l) * F_ + i) * D_;
    #pragma unroll 4
    for (int kt = 0; kt < D_ / 4; ++kt) {
        const int k0 = kt * 4 + half * 2;
        v2f a; a.x = arow[k0]; a.y = arow[k0 + 1];
        #pragma unroll
        for (int n = 0; n < 4; ++n) {
            const int col = n * 16 + l;
            v2f bf; bf.x = sW[k0 * D_ + col]; bf.y = sW[(k0 + 1) * D_ + col];
            acc[n] = __builtin_amdgcn_wmma_f32_16x16x4_f32(
                false, a, false, bf, (short)0, acc[n], false, false);
        }
    }

    // C/D layout: acc[n][v] = proj[row = v + 8*half, col = n*16 + l]
    const float* __restrict__ bprow = bp + (size_t)p * D_;
    #pragma unroll
    for (int v = 0; v < 8; ++v) {
        const int row = mbase + v + half * 8;
        const float* __restrict__ xjrow = x + ((size_t)row * F_ + j) * D_;
        float s = 0.f;
        #pragma unroll
        for (int n = 0; n < 4; ++n) {
            const int col = n * 16 + l;
            s += (acc[n][v] + bprow[col]) * xjrow[col];
        }
        s += __shfl_xor(s, 1, 32);
        s += __shfl_xor(s, 2, 32);
        s += __shfl_xor(s, 4, 32);
        s += __shfl_xor(s, 8, 32);
        if (l == 0) dots[(size_t)row * P_ + p] = s;
    }
}

// ---------------------------------------------------------------------------
// Kernel 2: grid = (P_/2, B_/64), block = 128 threads (4 waves).
// Each block handles TWO pairs with double-buffered TDM staging of Wp tiles:
// wave 0 issues both tensor_load_to_lds up front (per-wave TDM ops complete
// in order), compute on pair 0 overlaps the in-flight load of pair 1.
// ---------------------------------------------------------------------------
__global__ __launch_bounds__(128) void pair_proj_dots_kernel(
    const float* __restrict__ x, const float* __restrict__ Wp,
    const float* __restrict__ bp, float* __restrict__ dots)
{
    __shared__ float sW[2][D_ * D_];   // 2 x 16 KB double buffer

    const int t = threadIdx.x;
    const int wave = t >> 5, lane = t & 31;
    const int half = lane >> 4, l = lane & 15;
    const int mbase = (blockIdx.y * 4 + wave) * 16;

    const int p0 = blockIdx.x * 2;
    const int p1 = p0 + 1;

    if (wave == 0) {
        tdm_load_tile_64x64_f32(&sW[0][0], Wp + (size_t)p0 * (D_ * D_));
        tdm_load_tile_64x64_f32(&sW[1][0], Wp + (size_t)p1 * (D_ * D_));
    }

    // unrank p0 -> (i0, j0); p1 is the next lexicographic pair
    int i0 = 0, rem = p0;
    while (rem >= F_ - 1 - i0) { rem -= F_ - 1 - i0; ++i0; }
    int j0 = i0 + 1 + rem;
    int i1 = i0, j1 = j0 + 1;
    if (j1 >= F_) { i1 = i0 + 1; j1 = i1 + 1; }

    // ---- pair 0: wait for first TDM op only (second may still be in flight)
    if (wave == 0) __builtin_amdgcn_s_wait_tensorcnt(1);
    __syncthreads();
    pair_tile_compute(x, &sW[0][0], bp, dots, p0, i0, j0, mbase, half, l);

    // ---- pair 1
    if (wave == 0) __builtin_amdgcn_s_wait_tensorcnt(0);
    __syncthreads();
    pair_tile_compute(x, &sW[1][0], bp, dots, p1, i1, j1, mbase, half, l);
}

// ---------------------------------------------------------------------------
// Kernel 3: bilinear = dots(1024x496) @ Wo(496x512) + bo -> out[:, 2048:2560]
// One wave per 16x16 output tile; K=496 -> 124 fp32 WMMAs per tile.
// ---------------------------------------------------------------------------
__global__ __launch_bounds__(128) void bilinear_out_kernel(
    const float* __restrict__ dots, const float* __restrict__ Wo,
    const float* __restrict__ bo, float* __restrict__ out)
{
    const int t = threadIdx.x;
    const int wave = t >> 5, lane = t & 31;
    const int half = lane >> 4, l = lane & 15;
    const int mbase = blockIdx.x * 16;
    const int nbase = (blockIdx.y * 4 + wave) * 16;

    const v8f vzero = {0.f, 0.f, 0.f, 0.f, 0.f, 0.f, 0.f, 0.f};
    v8f acc = vzero;

    const float* __restrict__ arow = dots + (size_t)(mbase + l) * P_;
    const int col = nbase + l;
    #pragma unroll 4
    for (int kt = 0; kt < P_ / 4; ++kt) {
        const int k0 = kt * 4 + half * 2;
        v2f a;  a.x  = arow[k0];                    a.y  = arow[k0 + 1];
        v2f bf; bf.x = Wo[(size_t)k0 * OUT_ + col]; bf.y = Wo[(size_t)(k0 + 1) * OUT_ + col];
        acc = __builtin_amdgcn_wmma_f32_16x16x4_f32(
            false, a, false, bf, (short)0, acc, false, false);
    }

    const float bov = bo[col];
    #pragma unroll
    for (int v = 0; v < 8; ++v) {
        const int row = mbase + v + half * 8;
        out[(size_t)row * OSTRIDE_ + EMB_ + col] = acc[v] + bov;
    }
}

// ---------------------------------------------------------------------------
extern "C" void kernel_launch(void* const* d_in, const int* in_sizes, int n_in,
                              void* d_out, int out_size, void* d_ws, size_t ws_size,
                              hipStream_t stream)
{
    (void)in_sizes; (void)n_in; (void)out_size; (void)ws_size;
    const float* x        = (const float*)d_in[0];
    const float* W1       = (const float*)d_in[1];
    const float* b1       = (const float*)d_in[2];
    const float* W2       = (const float*)d_in[3];
    const float* b2       = (const float*)d_in[4];
    const float* ln_scale = (const float*)d_in[5];
    const float* ln_bias  = (const float*)d_in[6];
    const float* Wp       = (const float*)d_in[7];
    const float* bp       = (const float*)d_in[8];
    const float* Wo       = (const float*)d_in[9];
    const float* bo       = (const float*)d_in[10];
    float* out  = (float*)d_out;
    float* dots = (float*)d_ws;   // B_*P_ floats (~2 MB scratch)

    senet_ln_kernel<<<dim3(B_), dim3(256), 0, stream>>>(
        x, W1, b1, W2, b2, ln_scale, ln_bias, out);
    pair_proj_dots_kernel<<<dim3(P_ / 2, B_ / 64), dim3(128), 0, stream>>>(
        x, Wp, bp, dots);
    bilinear_out_kernel<<<dim3(B_ / 16, OUT_ / 64), dim3(128), 0, stream>>>(
        dots, Wo, bo, out);
}